// PGNDecoder_30391188587359
// MI455X (gfx1250) — compile-verified
//
#include <hip/hip_runtime.h>
#include <math.h>

// Problem constants (from reference)
#define V_    50000
#define EMB_  128
#define H_    512
#define ENC_  512
#define B_    64
#define S_    400
#define KX    (EMB_ + ENC_)      // 640  : LSTM input x = [emb | prev_context]
#define G4H   (4 * H_)           // 2048 : gates
#define KV    (ENC_ + H_)        // 1024 : vocab-proj input [context | h]
#define PGK   (ENC_ + H_ + EMB_) // 1152 : p_gen input
#define KCHUNK 128               // TDM chunk rows for Wv streaming

typedef float v2f __attribute__((ext_vector_type(2)));
typedef float v8f __attribute__((ext_vector_type(8)));
typedef unsigned int v4u __attribute__((ext_vector_type(4)));
typedef int v4i __attribute__((ext_vector_type(4)));
typedef int v8i __attribute__((ext_vector_type(8)));

// hardware rcp (v_rcp_f32) instead of IEEE div sequence
__device__ __forceinline__ float frcp(float x) { return __builtin_amdgcn_rcpf(x); }
__device__ __forceinline__ float sigm(float x) { return frcp(1.0f + __expf(-x)); }
// branch-free tanh: 1 - 2/(e^{2x}+1)  (v_exp_f32 + v_rcp_f32, no EXEC branching)
__device__ __forceinline__ float ftanh(float x) {
  float e = __expf(2.0f * x);
  return 1.0f - 2.0f * frcp(e + 1.0f);
}

// ---------------------------------------------------------------------------
// Kernel 1: build x = [emb(input_ids) | prev_context]  (64 x 640)
// ---------------------------------------------------------------------------
__global__ void pack_x_kernel(const int* __restrict__ ids,
                              const float* __restrict__ emb_table,
                              const float* __restrict__ prev_ctx,
                              float* __restrict__ x) {
  int idx = blockIdx.x * blockDim.x + threadIdx.x;
  if (idx >= B_ * KX) return;
  int b = idx / KX, k = idx % KX;
  float v;
  if (k < EMB_) v = emb_table[(size_t)ids[b] * EMB_ + k];
  else          v = prev_ctx[b * ENC_ + (k - EMB_)];
  x[idx] = v;
}

// ---------------------------------------------------------------------------
// Generic M=64 fp32 WMMA GEMM: C[64,N] (+)= A[64,K] @ W[K,N] (+ bias0 + bias1)
// grid.x = N/16, block = 128 (4 waves; wave w owns rows [16w, 16w+16)).
// K, N compile-time so the stride-N load folds into an immediate offset and
// the loop body is pure {2 loads, 1 wmma, 2 pointer bumps}.
// Fragment layout per CDNA5 ISA 7.12.2 (16x16x4 f32).
// ---------------------------------------------------------------------------
template <int K, int N, bool ACC>
__global__ void gemm64_wmma(const float* __restrict__ A, const float* __restrict__ W,
                            const float* __restrict__ bias0, const float* __restrict__ bias1,
                            float* __restrict__ C) {
  const int lane = threadIdx.x & 31;
  const int wave = threadIdx.x >> 5;       // 0..3 -> M-tile
  const int n0   = blockIdx.x * 16;
  const int l15  = lane & 15;
  const int hi   = (lane >= 16) ? 1 : 0;
  const int koff = hi * 2;
  const int m0   = wave * 16;

  v8f acc = {};
  if (ACC) {
#pragma unroll
    for (int j = 0; j < 8; ++j) {
      int row = m0 + j + hi * 8;
      acc[j] = C[(size_t)row * N + n0 + l15];
    }
  }

  const float* pA = A + (size_t)(m0 + l15) * K + koff;   // bump +4 floats / iter
  const float* pW = W + (size_t)koff * N + n0 + l15;     // bump +4N floats / iter
#pragma unroll 4
  for (int k0 = 0; k0 < K; k0 += 4) {
    const float2 av = *(const float2*)pA;
    v2f a; a.x = av.x; a.y = av.y;
    v2f b; b.x = pW[0]; b.y = pW[N];                     // imm offset N*4 bytes
    acc = __builtin_amdgcn_wmma_f32_16x16x4_f32(false, a, false, b, (short)0, acc,
                                                false, false);
    pA += 4;
    pW += 4 * N;
  }

#pragma unroll
  for (int j = 0; j < 8; ++j) {
    int row = m0 + j + hi * 8;
    float val = acc[j];
    if (bias0) val += bias0[n0 + l15];
    if (bias1) val += bias1[n0 + l15];
    C[(size_t)row * N + n0 + l15] = val;
  }
}

// ---------------------------------------------------------------------------
// Kernel 3: LSTM elementwise (torch gate order i,f,g,o)
// ---------------------------------------------------------------------------
__global__ void lstm_elem_kernel(const float* __restrict__ gates,
                                 const float* __restrict__ c0,
                                 float* __restrict__ h_ws,
                                 float* __restrict__ h_out,
                                 float* __restrict__ c_out) {
  int idx = blockIdx.x * blockDim.x + threadIdx.x;
  if (idx >= B_ * H_) return;
  int b = idx / H_, d = idx % H_;
  const float* g = gates + (size_t)b * G4H;
  float gi = g[d], gf = g[H_ + d], gg = g[2 * H_ + d], go = g[3 * H_ + d];
  float c = sigm(gf) * c0[idx] + sigm(gi) * ftanh(gg);
  float h = sigm(go) * ftanh(c);
  h_ws[idx] = h; h_out[idx] = h; c_out[idx] = c;
}

// ---------------------------------------------------------------------------
// Kernel 5: attention scores.  Per block: one (b, 16-row s-tile).
// The 16x512 A tile (enc rows) is staged once into LDS in transposed [k][m]
// layout (coalesced global reads; conflict-free ds reads: lanes 0-15 hit 16
// consecutive banks, lanes 16-31 the +32 group).  enc_feat via f32 WMMA,
// fused + dec_feat + cov*Wc, tanh, . v.  Per-lane partials across the 4
// n-tiles, committed with 8 ds_add_f32 per lane at the end.
// ---------------------------------------------------------------------------
__global__ void attn_scores_kernel(const float* __restrict__ enc,
                                   const float* __restrict__ Wh,
                                   const float* __restrict__ decfeat,
                                   const float* __restrict__ cov,
                                   const float* __restrict__ Wc,
                                   const float* __restrict__ vvec,
                                   const int* __restrict__ mask,
                                   float* __restrict__ scores) {
  __shared__ float sA[ENC_ * 16];          // [k][m] transposed tile, 32 KB
  __shared__ float sscore[16];
  const int b    = blockIdx.y;
  const int s0   = blockIdx.x * 16;
  const int lane = threadIdx.x & 31;
  const int wave = threadIdx.x >> 5;       // 8 waves
  const int l15  = lane & 15;
  const int hi   = (lane >= 16) ? 1 : 0;
  const int koff = hi * 2;

  // stage A tile: coalesced along k, transposed into LDS
  for (int idx = threadIdx.x; idx < 16 * ENC_; idx += 256) {
    int m = idx >> 9;                      // idx / 512
    int k = idx & (ENC_ - 1);
    sA[k * 16 + m] = enc[(size_t)(b * S_ + s0 + m) * ENC_ + k];
  }
  if (threadIdx.x < 16) sscore[threadIdx.x] = 0.0f;
  __syncthreads();

  // coverage per output row (same for all n-tiles)
  const int mrow = hi * 8;
  float cvv[8];
#pragma unroll
  for (int j = 0; j < 8; ++j) cvv[j] = cov[b * S_ + s0 + mrow + j];

  float part[8] = {};
  for (int nt = wave; nt < H_ / 16; nt += 8) {
    const int n0 = nt * 16;
    v8f acc = {};
    const float* pW = Wh + (size_t)koff * H_ + n0 + l15;
#pragma unroll 4
    for (int k0 = 0; k0 < ENC_; k0 += 4) {
      v2f a;
      a.x = sA[(k0 + koff) * 16 + l15];
      a.y = sA[(k0 + koff + 1) * 16 + l15];
      v2f bb; bb.x = pW[0]; bb.y = pW[H_];
      acc = __builtin_amdgcn_wmma_f32_16x16x4_f32(false, a, false, bb, (short)0, acc,
                                                  false, false);
      pW += 4 * H_;
    }
    const int ncol = n0 + l15;
    const float dn  = decfeat[b * H_ + ncol];
    const float wcn = Wc[ncol];
    const float vn  = vvec[ncol];
#pragma unroll
    for (int j = 0; j < 8; ++j)
      part[j] += ftanh(acc[j] + dn + cvv[j] * wcn) * vn;
  }
#pragma unroll
  for (int j = 0; j < 8; ++j)
    atomicAdd(&sscore[mrow + j], part[j]);               // ds_add_f32
  __syncthreads();
  if (threadIdx.x < 16) {
    int s = s0 + threadIdx.x;
    float sc = sscore[threadIdx.x];
    if (mask[b * S_ + s] == 0) sc = -1e9f;
    scores[b * S_ + s] = sc;
  }
}

// ---------------------------------------------------------------------------
// Kernel 6: softmax over S per row + context = attn @ enc_outputs
// ---------------------------------------------------------------------------
__global__ void attn_softmax_context(const float* __restrict__ scores,
                                     const float* __restrict__ enc,
                                     float* __restrict__ attn_out,
                                     float* __restrict__ ctx_ws,
                                     float* __restrict__ ctx_out) {
  __shared__ float sbuf[S_];
  __shared__ float red[256];
  const int b = blockIdx.x;
  const int tid = threadIdx.x;
  float lmax = -3.4e38f;
  for (int s = tid; s < S_; s += 256) {
    float v = scores[b * S_ + s]; sbuf[s] = v; lmax = fmaxf(lmax, v);
  }
  red[tid] = lmax; __syncthreads();
  for (int off = 128; off > 0; off >>= 1) {
    if (tid < off) red[tid] = fmaxf(red[tid], red[tid + off]);
    __syncthreads();
  }
  float rmax = red[0]; __syncthreads();
  float lsum = 0.f;
  for (int s = tid; s < S_; s += 256) {
    float e = __expf(sbuf[s] - rmax); sbuf[s] = e; lsum += e;
  }
  red[tid] = lsum; __syncthreads();
  for (int off = 128; off > 0; off >>= 1) {
    if (tid < off) red[tid] += red[tid + off];
    __syncthreads();
  }
  float rinv = 1.0f / red[0]; __syncthreads();
  for (int s = tid; s < S_; s += 256) {
    float a = sbuf[s] * rinv; sbuf[s] = a; attn_out[b * S_ + s] = a;
  }
  __syncthreads();
  for (int d = tid; d < ENC_; d += 256) {
    float acc = 0.f;
    const float* ep = enc + (size_t)b * S_ * ENC_ + d;
    for (int s = 0; s < S_; ++s) acc += sbuf[s] * ep[(size_t)s * ENC_];
    ctx_ws[b * ENC_ + d] = acc;
    ctx_out[b * ENC_ + d] = acc;
  }
}

// ---------------------------------------------------------------------------
// Kernel 7: pack xv = [context | h] and compute p_gen
// ---------------------------------------------------------------------------
__global__ void pg_pack_kernel(const float* __restrict__ ctx,
                               const float* __restrict__ h,
                               const float* __restrict__ x,
                               const float* __restrict__ Wpg,
                               const float* __restrict__ bpg,
                               const float* __restrict__ pgb,
                               float* __restrict__ xv,
                               float* __restrict__ pgen_out) {
  __shared__ float red[256];
  const int b = blockIdx.x; const int tid = threadIdx.x;
  for (int i = tid; i < KV; i += 256) {
    float v = (i < ENC_) ? ctx[b * ENC_ + i] : h[b * H_ + (i - ENC_)];
    xv[(size_t)b * KV + i] = v;
  }
  float part = 0.f;
  for (int i = tid; i < PGK; i += 256) {
    float v;
    if (i < ENC_)            v = ctx[b * ENC_ + i];
    else if (i < ENC_ + H_)  v = h[b * H_ + (i - ENC_)];
    else                     v = x[b * KX + (i - ENC_ - H_)];  // emb lives at x[:,0:128]
    part += v * Wpg[i];
  }
  red[tid] = part; __syncthreads();
  for (int off = 128; off > 0; off >>= 1) {
    if (tid < off) red[tid] += red[tid + off];
    __syncthreads();
  }
  if (tid == 0) {
    float p = sigm(red[0] + bpg[0] + pgb[0]);
    pgen_out[b] = fminf(0.9f, fmaxf(0.1f, p));
  }
}

// ---------------------------------------------------------------------------
// Kernel 8: vocab logits  [64 x 1024] @ Wv[1024 x 50000] + bv.
// HBM-bound: streams the 205 MB Wv exactly once.  The Tensor Data Mover
// double-buffers each 16-col x 128-row Wv tile into LDS (D#: 2D tensor,
// data_size=4B, tensor_dim0_stride=V) while the wave runs WMMA on the
// previous tile.  TENSORcnt is per-wave & in-order, so issue(c+1) then
// s_wait_tensorcnt(1) guarantees chunk c is resident.  One wave per block.
// ---------------------------------------------------------------------------
__device__ __forceinline__ void tdm_issue_tile(const float* gsrc, unsigned lds_off,
                                               const v8i& g1) {
  unsigned long long ga = (unsigned long long)(uintptr_t)gsrc;
  v4u g0;
  g0[0] = 1u;                                             // count=1, user mode
  g0[1] = lds_off;                                        // LDS byte address
  g0[2] = (unsigned)ga;                                   // global_addr[31:0]
  g0[3] = (unsigned)((ga >> 32) & 0x01FFFFFFu) | (2u << 30); // addr[56:32] | type=2
  v4i gz = {0, 0, 0, 0};
#if __clang_major__ >= 23
  v8i gz8 = {0, 0, 0, 0, 0, 0, 0, 0};
  __builtin_amdgcn_tensor_load_to_lds(g0, g1, gz, gz, gz8, 0);
#else
  __builtin_amdgcn_tensor_load_to_lds(g0, g1, gz, gz, 0);
#endif
}

__global__ void vocab_gemm_wmma(const float* __restrict__ xv,
                                const float* __restrict__ Wv,
                                const float* __restrict__ bv,
                                float* __restrict__ logits) {
  __shared__ float sB[2][KCHUNK * 16];
  const int lane = threadIdx.x & 31;
  const int n0   = blockIdx.x * 16;
  const int l15  = lane & 15;
  const int hi   = (lane >= 16) ? 1 : 0;
  const int koff = hi * 2;

  // D# group1 (wave-uniform): tensor 1024 x 50000, tile 128 x 16, 4B elems
  v8i g1;
  g1[0] = 0x00020000;                                    // data_size = 4B
  g1[1] = (int)((unsigned)(V_ & 0xFFFF) << 16);          // tensor_dim0[15:0]
  g1[2] = (int)(((V_ >> 16) & 0xFFFF) | ((KV & 0xFFFF) << 16)); // dim0 hi | dim1 lo
  g1[3] = (int)(((KV >> 16) & 0xFFFF) | (16u << 16));    // dim1 hi | tile_dim0=16
  g1[4] = KCHUNK;                                        // tile_dim1 | tile_dim2=0
  g1[5] = V_;                                            // tensor_dim0_stride[31:0]
  g1[6] = 0;
  g1[7] = 0;

  const unsigned lds0 = (unsigned)(uintptr_t)&sB[0][0];
  const unsigned lds1 = (unsigned)(uintptr_t)&sB[1][0];

  v8f acc[4] = {};
  const float* pA0 = xv + (size_t)l15 * KV + koff;

  tdm_issue_tile(Wv + (size_t)0 * V_ + n0, lds0, g1);
  for (int c = 0; c < KV / KCHUNK; ++c) {
    if (c + 1 < KV / KCHUNK) {
      tdm_issue_tile(Wv + (size_t)((c + 1) * KCHUNK) * V_ + n0,
                     (c & 1) ? lds0 : lds1, g1);
      __builtin_amdgcn_s_wait_tensorcnt(1);
    } else {
      __builtin_amdgcn_s_wait_tensorcnt(0);
    }
    const int buf = c & 1;
    const float* pA = pA0 + c * KCHUNK;
#pragma unroll 4
    for (int kk = 0; kk < KCHUNK; kk += 4) {
      v2f bb;
      bb.x = sB[buf][(kk + koff) * 16 + l15];            // ds_load, conflict-free
      bb.y = sB[buf][(kk + koff + 1) * 16 + l15];
#pragma unroll
      for (int m = 0; m < 4; ++m) {
        const float2 av = *(const float2*)(pA + (size_t)m * 16 * KV + kk);
        v2f a; a.x = av.x; a.y = av.y;
        acc[m] = __builtin_amdgcn_wmma_f32_16x16x4_f32(false, a, false, bb, (short)0,
                                                       acc[m], false, false);
      }
    }
  }

  const float bias = bv[n0 + l15];
#pragma unroll
  for (int m = 0; m < 4; ++m) {
#pragma unroll
    for (int j = 0; j < 8; ++j) {
      int row = m * 16 + j + hi * 8;
      logits[(size_t)row * V_ + n0 + l15] = acc[m][j] + bias;
    }
  }
}

// ---------------------------------------------------------------------------
// Kernel 9: softmax over V per row, in place on d_out vocab region
// ---------------------------------------------------------------------------
__global__ void vocab_softmax_kernel(float* __restrict__ logits) {
  __shared__ float red[256];
  const int b = blockIdx.x; const int tid = threadIdx.x;
  float* row = logits + (size_t)b * V_;
  float lmax = -3.4e38f;
  for (int i = tid; i < V_; i += 256) lmax = fmaxf(lmax, row[i]);
  red[tid] = lmax; __syncthreads();
  for (int off = 128; off > 0; off >>= 1) {
    if (tid < off) red[tid] = fmaxf(red[tid], red[tid + off]);
    __syncthreads();
  }
  float rmax = red[0]; __syncthreads();
  float lsum = 0.f;
  for (int i = tid; i < V_; i += 256) lsum += __expf(row[i] - rmax);
  red[tid] = lsum; __syncthreads();
  for (int off = 128; off > 0; off >>= 1) {
    if (tid < off) red[tid] += red[tid + off];
    __syncthreads();
  }
  float rinv = 1.0f / red[0]; __syncthreads();
  for (int i = tid; i < V_; i += 256) row[i] = __expf(row[i] - rmax) * rinv;
}

// ---------------------------------------------------------------------------
extern "C" void kernel_launch(void* const* d_in, const int* in_sizes, int n_in,
                              void* d_out, int out_size, void* d_ws, size_t ws_size,
                              hipStream_t stream) {
  const int*   ids      = (const int*)  d_in[0];
  const float* h0       = (const float*)d_in[1];
  const float* c0       = (const float*)d_in[2];
  const float* enc      = (const float*)d_in[3];
  const int*   enc_mask = (const int*)  d_in[4];
  const float* coverage = (const float*)d_in[5];
  const float* prev_ctx = (const float*)d_in[6];
  const float* pgb      = (const float*)d_in[7];
  const float* W_ih     = (const float*)d_in[8];
  const float* W_hh     = (const float*)d_in[9];
  const float* b_ih     = (const float*)d_in[10];
  const float* b_hh     = (const float*)d_in[11];
  const float* Wh       = (const float*)d_in[12];
  const float* Ws       = (const float*)d_in[13];
  const float* bs       = (const float*)d_in[14];
  const float* Wc       = (const float*)d_in[15];
  const float* vvec     = (const float*)d_in[16];
  const float* Wpg      = (const float*)d_in[17];
  const float* bpg      = (const float*)d_in[18];
  const float* Wv       = (const float*)d_in[19];
  const float* bv       = (const float*)d_in[20];
  const float* emb_tab  = (const float*)d_in[21];

  // d_out layout: vocab_dist, attn, p_gen, h, c, context (flat, in return order)
  float* out = (float*)d_out;
  float* out_vocab = out;                          // 64*50000
  float* out_attn  = out_vocab + (size_t)B_ * V_;  // 64*400
  float* out_pg    = out_attn  + (size_t)B_ * S_;  // 64
  float* out_h     = out_pg    + B_;               // 64*512
  float* out_c     = out_h     + (size_t)B_ * H_;  // 64*512
  float* out_ctx   = out_c     + (size_t)B_ * H_;  // 64*512

  // workspace layout (floats)
  float* ws      = (float*)d_ws;
  float* w_x     = ws;                                  // 64*640
  float* w_gates = w_x     + (size_t)B_ * KX;           // 64*2048
  float* w_h     = w_gates + (size_t)B_ * G4H;          // 64*512
  float* w_dec   = w_h     + (size_t)B_ * H_;           // 64*512
  float* w_sc    = w_dec   + (size_t)B_ * H_;           // 64*400
  float* w_ctx   = w_sc    + (size_t)B_ * S_;           // 64*512
  float* w_xv    = w_ctx   + (size_t)B_ * ENC_;         // 64*1024

  // 1. x = [emb | prev_context]
  pack_x_kernel<<<(B_ * KX + 255) / 256, 256, 0, stream>>>(ids, emb_tab, prev_ctx, w_x);

  // 2. gates = x @ W_ih + b_ih + b_hh ; then += h0 @ W_hh
  gemm64_wmma<KX, G4H, false><<<G4H / 16, 128, 0, stream>>>(w_x, W_ih, b_ih, b_hh, w_gates);
  gemm64_wmma<H_, G4H, true ><<<G4H / 16, 128, 0, stream>>>(h0, W_hh, nullptr, nullptr, w_gates);

  // 3. LSTM elementwise -> h, c
  lstm_elem_kernel<<<(B_ * H_ + 255) / 256, 256, 0, stream>>>(w_gates, c0, w_h, out_h, out_c);

  // 4. dec_feat = h @ Ws + bs
  gemm64_wmma<H_, H_, false><<<H_ / 16, 128, 0, stream>>>(w_h, Ws, bs, nullptr, w_dec);

  // 5. attention scores (WMMA GEMM fused with tanh / v-dot)
  attn_scores_kernel<<<dim3(S_ / 16, B_), 256, 0, stream>>>(
      enc, Wh, w_dec, coverage, Wc, vvec, enc_mask, w_sc);

  // 6. softmax over S + context
  attn_softmax_context<<<B_, 256, 0, stream>>>(w_sc, enc, out_attn, w_ctx, out_ctx);

  // 7. xv = [context | h]; p_gen
  pg_pack_kernel<<<B_, 256, 0, stream>>>(w_ctx, w_h, w_x, Wpg, bpg, pgb, w_xv, out_pg);

  // 8. vocab logits (dominant: TDM-streams Wv 205 MB once, double-buffered)
  vocab_gemm_wmma<<<V_ / 16, 32, 0, stream>>>(w_xv, Wv, bv, out_vocab);

  // 9. softmax over V in place
  vocab_softmax_kernel<<<B_, 256, 0, stream>>>(out_vocab);
}